// SetKernel_21260088115410
// MI455X (gfx1250) — compile-verified
//
#include <hip/hip_runtime.h>
#include <hip/hip_bf16.h>

typedef __attribute__((ext_vector_type(2))) float v2f;
typedef __attribute__((ext_vector_type(8))) float v8f;

#define NX 8192
#define NY 8192
#define SX 2048

// ---------------------------------------------------------------------------
// Phase 1: row means of x (8192x2048) and y (8192x2048) -> means[16384].
// One 256-thread block per row; float4 loads (global_load_b128), LDS tree
// reduction. Pure streaming: 128 MiB reads dominate (~6 us at 23.3 TB/s).
// ---------------------------------------------------------------------------
__global__ void row_mean_kernel(const float* __restrict__ x,
                                const float* __restrict__ y,
                                float* __restrict__ means) {
  const int row = blockIdx.x;  // 0..16383; first 8192 are x rows
  const float* src = (row < NX) ? (x + (size_t)row * SX)
                                : (y + (size_t)(row - NX) * SX);
  const float4* src4 = (const float4*)src;

  float s = 0.0f;
  for (int i = threadIdx.x; i < SX / 4; i += 256) {
    float4 v = src4[i];
    s += (v.x + v.y) + (v.z + v.w);
  }

  __shared__ float red[256];
  red[threadIdx.x] = s;
  __syncthreads();
  #pragma unroll
  for (int off = 128; off > 0; off >>= 1) {
    if (threadIdx.x < off) red[threadIdx.x] += red[threadIdx.x + off];
    __syncthreads();
  }
  if (threadIdx.x == 0) means[row] = red[0] * (1.0f / (float)SX);
}

// ---------------------------------------------------------------------------
// Phase 2: out[n,m] = exp(-(mX[n]-mY[m])^2 / (2 sigma^2)).
// (mX-mY)^2 = mX^2*1 + mX*(-2 mY) + 1*mY^2 + 0  ==  K=4 contraction, so one
// V_WMMA_F32_16X16X4_F32 yields a full 16x16 FK tile in fp32.
//
// 32-bit A 16x4 layout (wave32): lanes 0-15 hold M=lane, VGPR{0,1} = K{0,1};
// lanes 16-31 hold M=lane-16, VGPR{0,1} = K{2,3}. B 4x16 is the mirror.
// Using the same slot convention for A and B makes sum_k A_k*B_k invariant
// under any common K permutation, so the construction is layout-robust.
//
// D layout: VGPR r, lanes 0-15 -> (row=r, col=lane); lanes 16-31 ->
// (row=r+8, col=lane-16). Each wave owns one 16x16 tile; 8 waves per block
// cover 128 consecutive m columns (coalesced streaming stores, 256 MB total).
//
// Scale factor without any divide:
//   sigma = e^ls  =>  -log2(e)/(2 sigma^2) = -log2(e)/2 * 2^(-2*ls*log2(e))
// Raw v_exp_f32 (__builtin_amdgcn_exp2f) is used for all exponentials; its
// flush-to-zero for very negative inputs matches exp(-huge) -> 0.
// ---------------------------------------------------------------------------
__global__ void gram_exp_kernel(const float* __restrict__ means,
                                const float* __restrict__ log_sigma,
                                float* __restrict__ out) {
  const float* mX = means;        // [NX]
  const float* mY = means + NX;   // [NY]

  const int lane = threadIdx.x & 31;
  const int wv   = threadIdx.x >> 5;               // wave id in block: 0..7
  const int n0   = blockIdx.y << 4;                // tile row base (x index)
  const int m0   = ((blockIdx.x << 3) + wv) << 4;  // tile col base (y index)
  const int col  = lane & 15;
  const bool hi  = lane >= 16;

  const float mx = mX[n0 + col];  // lane holds M = col for A
  const float my = mY[m0 + col];  // lane holds N = col for B

  // A row (per M): K0 = mx^2, K1 = mx, K2 = 1, K3 = 0
  v2f a;
  a.x = hi ? 1.0f : mx * mx;
  a.y = hi ? 0.0f : mx;
  // B col (per N): K0 = 1, K1 = -2*my, K2 = my^2, K3 = 0
  v2f b;
  b.x = hi ? my * my : 1.0f;
  b.y = hi ? 0.0f : -2.0f * my;

  v8f c = {};
  v8f d = __builtin_amdgcn_wmma_f32_16x16x4_f32(
      /*neg_a=*/false, a, /*neg_b=*/false, b,
      /*c_mod=*/(short)0, c, /*reuse_a=*/false, /*reuse_b=*/false);

  const float ls = log_sigma[0];
  // cc = -log2(e)/2 * 2^(-2*ls*log2(e))  ==  -log2(e) / (2*e^(2*ls))
  const float cc = -0.7213475204444817f *
                   __builtin_amdgcn_exp2f(ls * -2.8853900817779268f);

  float* outp = out + (size_t)(n0 + (hi ? 8 : 0)) * (size_t)NY + (size_t)(m0 + col);
  #pragma unroll
  for (int r = 0; r < 8; ++r) {
    outp[(size_t)r * (size_t)NY] = __builtin_amdgcn_exp2f(d[r] * cc);
  }
}

extern "C" void kernel_launch(void* const* d_in, const int* in_sizes, int n_in,
                              void* d_out, int out_size, void* d_ws, size_t ws_size,
                              hipStream_t stream) {
  const float* x         = (const float*)d_in[0];
  const float* y         = (const float*)d_in[1];
  const float* log_sigma = (const float*)d_in[2];
  float* out   = (float*)d_out;
  float* means = (float*)d_ws;  // 16384 floats = 64 KB scratch

  (void)in_sizes; (void)n_in; (void)out_size; (void)ws_size;

  // Phase 1: 16384 row reductions (x rows then y rows).
  row_mean_kernel<<<NX + NY, 256, 0, stream>>>(x, y, means);

  // Phase 2: 512x512 WMMA tiles of 16x16; 8 tiles (waves) per block.
  dim3 grid(NY / 128, NX / 16);
  gram_exp_kernel<<<grid, 256, 0, stream>>>(means, log_sigma, out);
}